// CapsuleLayer_40011915329898
// MI455X (gfx1250) — compile-verified
//
#include <hip/hip_runtime.h>

// CDNA5 / gfx1250: wave32, WMMA (not MFMA).
typedef float v2f __attribute__((ext_vector_type(2)));
typedef float v8f __attribute__((ext_vector_type(8)));

#define BB 64
#define RR 2048
#define CC 16
#define OO 32
#define II 16
#define CO 512   // C*O

// LDS layout (floats)
#define WS_OFF 0                    // W[r] tile: 512 x 16        = 8192
#define XS_OFF 8192                 // x[:,r,:]  : 64 x 16        = 1024
#define UH_OFF (8192 + 1024)        // u_hat     : 64 x 512       = 32768
#define PR_OFF (UH_OFF + 32768)     // probs     : 64 x 16        = 1024
#define VB_OFF (PR_OFF + 1024)      // v buffer  : 64 x 32        = 2048
#define SMEM_FLOATS (VB_OFF + 2048) // 45056 floats = 176 KB (< 320 KB/WGP)

// ---- CDNA5 async global->LDS staging via inline asm (ASYNCcnt path) ------
// ISA 15.18.3: GLOBAL_LOAD_ASYNC_TO_LDS_B128  vdst(LDS addr VGPR), vaddr, saddr
// Generic LDS pointer = {aperture_hi, lds_offset}: low 32 bits are the DS address.
__device__ __forceinline__ void async_load_b128(const void* gptr, void* lptr) {
    asm volatile("global_load_async_to_lds_b128 %0, %1, off"
                 :: "v"((uint32_t)(uintptr_t)lptr), "v"(gptr)
                 : "memory");
}
__device__ __forceinline__ void wait_async0() {
#if defined(__has_builtin)
# if __has_builtin(__builtin_amdgcn_s_wait_asynccnt)
    __builtin_amdgcn_s_wait_asynccnt(0);
# else
    asm volatile("s_wait_asynccnt 0x0" ::: "memory");
# endif
#else
    asm volatile("s_wait_asynccnt 0x0" ::: "memory");
#endif
}

__global__ __launch_bounds__(256)
void capsule_routing_kernel(const float* __restrict__ x,
                            const float* __restrict__ W,
                            float* __restrict__ out)
{
    extern __shared__ float smem[];
    float* ws = smem + WS_OFF;
    float* xs = smem + XS_OFF;
    float* uh = smem + UH_OFF;
    float* pr = smem + PR_OFF;
    float* vb = smem + VB_OFF;

    const int r    = blockIdx.x;
    const int tid  = threadIdx.x;
    const int lane = tid & 31;
    const int wave = tid >> 5;

    // ---------------- stage W[r] (512x16) and x[:,r,:] (64x16) into LDS ----
    // Direct memory->LDS DMA: no VGPR round-trip; tracked by ASYNCcnt.
    {
        const float4* Wg  = (const float4*)(W + (size_t)r * (CO * II));
        float4*       ws4 = (float4*)ws;
        #pragma unroll
        for (int j = 0; j < 8; ++j)                // 2048 float4 total
            async_load_b128(Wg + tid + j * 256, ws4 + tid + j * 256);

        const int bload = tid >> 2;                // one float4 of x per thread
        const int chunk = tid & 3;
        const float4* xg = (const float4*)(x + ((size_t)bload * RR + r) * II);
        async_load_b128(xg + chunk, ((float4*)xs) + tid);

        wait_async0();                             // drain this wave's ASYNCcnt
    }
    __syncthreads();                               // publish all waves' LDS writes

    // ---------------- u_hat = x(64x16) @ W[r]^T(16x512) via WMMA f32 ------
    // Tile map: 4 M-tiles (batch) x 32 N-tiles (c*32+o); 8 waves -> 16 tiles each.
    const int lrow   = lane & 15;   // M (for A/D) or N (for B/D) within tile
    const int khalf  = lane >> 4;   // selects K pair (A/B) / M+8 half (D)
    const int mt     = wave & 3;    // batch rows [mt*16, mt*16+16)
    const int ntbase = (wave >> 2) * 16;

    // A fragments (16x4 f32): VGPR0 = K{0|2}, VGPR1 = K{1|3} per ISA 7.12.2
    v2f afrag[4];
    #pragma unroll
    for (int ks = 0; ks < 4; ++ks) {
        const int k = ks * 4 + khalf * 2;
        afrag[ks].x = xs[(mt * 16 + lrow) * II + k];
        afrag[ks].y = xs[(mt * 16 + lrow) * II + k + 1];
    }

    #pragma unroll 1
    for (int nt = 0; nt < 16; ++nt) {
        const int n0 = (ntbase + nt) * 16;
        v8f acc = {};
        #pragma unroll
        for (int ks = 0; ks < 4; ++ks) {
            const int k = ks * 4 + khalf * 2;
            v2f bfrag;  // B (4x16 f32): row K striped across VGPRs, col N = lane&15
            bfrag.x = ws[(n0 + lrow) * II + k];
            bfrag.y = ws[(n0 + lrow) * II + k + 1];
            // (neg_a, A, neg_b, B, c_mod, C, reuse_a, reuse_b)
            acc = __builtin_amdgcn_wmma_f32_16x16x4_f32(
                false, afrag[ks], false, bfrag, (short)0, acc, false, false);
        }
        // D 16x16 f32: VGPR v -> M = v + 8*khalf, N = lrow
        #pragma unroll
        for (int v = 0; v < 8; ++v) {
            const int m = mt * 16 + v + khalf * 8;
            uh[m * CO + n0 + lrow] = acc[v];
        }
    }
    __syncthreads();

    // ---------------- dynamic routing: 3 iterations ------------------------
    // 4 lanes per batch row: b = tid>>2, sub = tid&3.
    //   sub owns capsules c in [sub*4, sub*4+4) and outputs o in [sub*8, sub*8+8).
    const int b   = tid >> 2;
    const int sub = tid & 3;
    float blog[4] = {0.f, 0.f, 0.f, 0.f};
    float vout[8];

    #pragma unroll 1
    for (int it = 0; it < 3; ++it) {
        // softmax over the 16 routing logits (quad reduction: xor 1,2 stays in quad)
        float mx = fmaxf(fmaxf(blog[0], blog[1]), fmaxf(blog[2], blog[3]));
        mx = fmaxf(mx, __shfl_xor(mx, 1));
        mx = fmaxf(mx, __shfl_xor(mx, 2));
        float e[4], esum = 0.f;
        #pragma unroll
        for (int j = 0; j < 4; ++j) { e[j] = __expf(blog[j] - mx); esum += e[j]; }
        esum += __shfl_xor(esum, 1);
        esum += __shfl_xor(esum, 2);
        const float inv = 1.0f / esum;
        #pragma unroll
        for (int j = 0; j < 4; ++j) pr[b * CC + sub * 4 + j] = e[j] * inv;
        __syncthreads();

        // s[o] = sum_c p[c] * u_hat[b,c,o]; then squash
        float p[16];
        #pragma unroll
        for (int c = 0; c < 16; ++c) p[c] = pr[b * CC + c];
        float s[8], nrm = 0.f;
        #pragma unroll
        for (int oo = 0; oo < 8; ++oo) {
            const int o = sub * 8 + oo;
            float a = 0.f;
            #pragma unroll
            for (int c = 0; c < 16; ++c) a += p[c] * uh[b * CO + c * OO + o];
            s[oo] = a;
            nrm  += a * a;
        }
        nrm += __shfl_xor(nrm, 1);
        nrm += __shfl_xor(nrm, 2);
        const float scale = nrm / ((1.f + nrm) * sqrtf(nrm));
        #pragma unroll
        for (int oo = 0; oo < 8; ++oo) {
            vout[oo] = scale * s[oo];
            vb[b * OO + sub * 8 + oo] = vout[oo];
        }
        __syncthreads();

        // agreement update: blog[c] += sum_o u_hat[b,c,o] * v[o] (skip last iter)
        if (it < 2) {
            float vv[32];
            #pragma unroll
            for (int o = 0; o < 32; ++o) vv[o] = vb[b * OO + o];
            #pragma unroll
            for (int j = 0; j < 4; ++j) {
                const int c = sub * 4 + j;
                float d = 0.f;
                #pragma unroll
                for (int o = 0; o < 32; ++o) d += uh[b * CO + c * OO + o] * vv[o];
                blog[j] += d;
            }
        }
        __syncthreads();
    }

    // ---------------- write v[b, r, o] -------------------------------------
    float* og = out + ((size_t)b * RR + r) * OO + sub * 8;
    #pragma unroll
    for (int oo = 0; oo < 8; ++oo) og[oo] = vout[oo];
}

extern "C" void kernel_launch(void* const* d_in, const int* in_sizes, int n_in,
                              void* d_out, int out_size, void* d_ws, size_t ws_size,
                              hipStream_t stream) {
    (void)in_sizes; (void)n_in; (void)d_ws; (void)ws_size; (void)out_size;
    const float* x = (const float*)d_in[0];   // [64, 2048, 16] fp32
    const float* W = (const float*)d_in[1];   // [2048, 16, 32, 16] fp32
    float* out = (float*)d_out;               // [64, 2048, 32] fp32

    const size_t smem = (size_t)SMEM_FLOATS * sizeof(float);  // 176 KB dynamic LDS
    (void)hipFuncSetAttribute((const void*)capsule_routing_kernel,
                              hipFuncAttributeMaxDynamicSharedMemorySize, (int)smem);
    capsule_routing_kernel<<<dim3(RR), dim3(256), smem, stream>>>(x, W, out);
}